// Kernel_21157008900103
// MI455X (gfx1250) — compile-verified
//
#include <hip/hip_runtime.h>

typedef float v2f __attribute__((ext_vector_type(2)));
typedef float v4f __attribute__((ext_vector_type(4)));
typedef float v8f __attribute__((ext_vector_type(8)));

#define NIN   576
#define NOUT  576
#define DIMY  25
#define NPATHR 77824
#define WPERZ 259
#define CBUF_FLOATS 2048   // wigner tensors use 1225, padded

// ---- static geometry tables (pair p = l_out*3 + l_in) ----
__constant__ int PAIR_WOFF[10]  = {0,1,4,9,12,39,84,89,134,259};
__constant__ int PATH_BASE[9]   = {0,1,2,3,4,7,10,11,14};
__constant__ int PATH_COFF[19]  = {0,1,10,35,44,53,80,125,170,245,350,375,420,495,600,625,700,825,1000};
__constant__ int PATH_PAIR[19]  = {0,1,2,3,4,4,4,5,5,5,6,7,7,7,8,8,8,8,8};
__constant__ int PATH_K[19]     = {0,0,0,0,0,1,2,0,1,2,0,0,1,2,0,1,2,3,4};

// constexpr copies for template kernels
__device__ constexpr int kPairWoff[9]   = {0,1,4,9,12,39,84,89,134};
__device__ constexpr int kPairBeginR[9] = {0,4096,8192,12288,16384,28672,40960,45056,57344};
__device__ constexpr int kBlkOff[3]     = {0,64,256};

// ======================= Wigner 3j on device (f64, one-time) ==================
struct C2 { double re, im; };
__device__ inline C2 cmul(C2 a, C2 b){ C2 r; r.re=a.re*b.re-a.im*b.im; r.im=a.re*b.im+a.im*b.re; return r; }
__device__ inline double dfact(int n){ double r=1.0; for(int i=2;i<=n;++i) r*=(double)i; return r; }
__device__ inline int imax2(int a,int b){return a>b?a:b;}
__device__ inline int imin2(int a,int b){return a<b?a:b;}

__device__ double su2_cg(int j1,int m1,int j2,int m2,int j3,int m3){
  if (m3 != m1+m2) return 0.0;
  int vmin = imax2(imax2(-j1+j2+m3, -j1+m1), 0);
  int vmax = imin2(imin2(j2+j3+m1, j3-j1+j2), j3+m3);
  double c = sqrt((double)(2*j3+1)*dfact(j3+j1-j2)*dfact(j3-j1+j2)*dfact(j1+j2-j3)/dfact(j1+j2+j3+1));
  c *= sqrt(dfact(j3+m3)*dfact(j3-m3)*dfact(j1-m1)*dfact(j1+m1)*dfact(j2-m2)*dfact(j2+m2));
  double s = 0.0;
  for (int v=vmin; v<=vmax; ++v) {
    s += (((v+j2+m2)&1)? -1.0:1.0)/dfact(v)
       * dfact(j2+j3+m1-v)*dfact(j1-m1+v)
       / dfact(j3-j1+j2-v)/dfact(j3+m3-v)/dfact(v+j1-j2-m3);
  }
  return c*s;
}

__device__ void build_q(int l, C2 q[9][9]) {
  int d = 2*l+1;
  for (int a=0;a<d;++a) for (int b=0;b<d;++b) { q[a][b].re=0.0; q[a][b].im=0.0; }
  const double is2 = 1.0/sqrt(2.0);
  for (int m=-l; m<0; ++m) {
    q[l+m][l-m].re = is2;             // column l+|m|
    q[l+m][l+m].im = -is2;            // column l-|m|
  }
  q[l][l].re = 1.0;
  for (int m=1; m<=l; ++m) {
    double s = (m&1)? -1.0 : 1.0;
    q[l+m][l+m].re = s*is2;
    q[l+m][l-m].im = s*is2;
  }
  C2 ph;                               // (-i)^l
  switch (l & 3) { case 0: ph={1,0}; break; case 1: ph={0,-1}; break;
                   case 2: ph={-1,0}; break; default: ph={0,1}; }
  for (int a=0;a<d;++a) for (int b=0;b<d;++b) q[a][b] = cmul(ph, q[a][b]);
}

__global__ void wigner_init_kernel(float* __restrict__ Cbuf) {
  int path = threadIdx.x;
  if (path >= 19) return;
  int p  = PATH_PAIR[path];
  int lo = p/3, li = p%3;
  int lf = (lo>li ? lo-li : li-lo) + PATH_K[path];
  int d1 = 2*lo+1, d2 = 2*li+1, d3 = 2*lf+1;

  double cg[5][5][9];
  for (int a=0;a<d1;++a) for (int b=0;b<d2;++b) for (int m=0;m<d3;++m) cg[a][b][m]=0.0;
  for (int m1=-lo; m1<=lo; ++m1)
    for (int m2=-li; m2<=li; ++m2) {
      int m3 = m1+m2;
      if (m3 >= -lf && m3 <= lf)
        cg[lo+m1][li+m2][lf+m3] = su2_cg(lo,m1,li,m2,lf,m3);
    }
  C2 q1[9][9], q2[9][9], q3[9][9];
  build_q(lo,q1); build_q(li,q2); build_q(lf,q3);

  double crv[5][5][9];
  double nrm2 = 0.0;
  for (int j=0;j<d1;++j) for (int l=0;l<d2;++l) for (int m=0;m<d3;++m) {
    double acc = 0.0;
    for (int i=0;i<d1;++i) {
      C2 a = q1[i][j];
      if (a.re==0.0 && a.im==0.0) continue;
      for (int k=0;k<d2;++k) {
        C2 b = q2[k][l];
        if (b.re==0.0 && b.im==0.0) continue;
        C2 ab = cmul(a,b);
        for (int n=0;n<d3;++n) {
          double cv = cg[i][k][n];
          if (cv==0.0) continue;
          acc += (ab.re*q3[n][m].re + ab.im*q3[n][m].im) * cv;   // Re(ab*conj(q3))*cv
        }
      }
    }
    crv[j][l][m] = acc;
    nrm2 += acc*acc;
  }
  double inv = 1.0/sqrt(nrm2);
  float* Cout = Cbuf + PATH_COFF[path];
  for (int j=0;j<d1;++j) for (int l=0;l<d2;++l) for (int m=0;m<d3;++m)
    Cout[(j*d2 + l)*d3 + m] = (float)(crv[j][l][m]*inv);
}

// ======================= W[z, pair, k, (i,j)] = nc * C·Y ======================
__global__ void w_kernel(const float* __restrict__ Y, const float* __restrict__ nc,
                         const float* __restrict__ Cbuf, float* __restrict__ Wbuf) {
  int z = blockIdx.x;
  for (int e = threadIdx.x; e < WPERZ; e += blockDim.x) {
    int p = 0;
    while (e >= PAIR_WOFF[p+1]) ++p;
    int lo = p/3, li = p%3;
    int P  = (2*lo+1)*(2*li+1);
    int local = e - PAIR_WOFF[p];
    int k  = local / P, ij = local % P;
    int lf = (lo>li ? lo-li : li-lo) + k;
    int dlf = 2*lf+1;
    const float* C  = Cbuf + PATH_COFF[PATH_BASE[p]+k] + ij*dlf;
    const float* Yz = Y + z*DIMY + lf*lf;
    float s = 0.f;
    for (int m = 0; m < dlf; ++m) s += C[m]*Yz[m];
    Wbuf[z*WPERZ + e] = nc[p] * s;
  }
}

// ============ helpers: unconditional clamped load + select (no exec branching) =
template<int NF_>
__device__ inline float rload(const float* __restrict__ p, int k) {
  float v = p[(k < NF_) ? k : 0];
  return (k < NF_) ? v : 0.0f;
}
template<int NF_, int P_>
__device__ inline float wload(const float* __restrict__ p, int k, int n) {
  int kk = (k < NF_) ? k : 0;
  int nn = (n < P_) ? n : 0;
  float v = p[kk*P_ + nn];
  return (k < NF_ && n < P_) ? v : 0.0f;
}

// ======================= main: WMMA GEMM + LDS transpose + NT coalesced store =
// one workgroup per (z, u); template per (l_out, l_in) pair (all shapes constexpr)
template<int LO, int LI>
__global__ __launch_bounds__(256) void tp_kernel(const float* __restrict__ R,
                                                 const float* __restrict__ Wbuf,
                                                 float* __restrict__ out) {
  constexpr int P      = (2*LO+1)*(2*LI+1);
  constexpr int NF     = 2*((LO<LI)?LO:LI)+1;
  constexpr int DIO    = 2*LO+1;
  constexpr int DII    = 2*LI+1;
  constexpr int NTILES = (P + 15) / 16;
  constexpr int ROWLEN = 64*DII;              // contiguous floats per (u,i) output row
  constexpr int CHUNKS = (DIO*ROWLEN) / 4;    // float4 chunks per block
  constexpr int PAIR   = LO*3 + LI;

  __shared__ float lds[64][P];                // P is odd -> conflict-benign stride

  int bid = blockIdx.x;
  int u = bid & 63;
  int z = bid >> 6;

  int tid  = threadIdx.x;
  int lane = tid & 31;
  int w    = tid >> 5;
  int vtile = w & 3, ntile = w >> 2;
  bool hi = lane >= 16;
  int l16 = lane & 15;

  const float* Rz = R    + (size_t)z * NPATHR + kPairBeginR[PAIR];
  const float* Wz = Wbuf + (size_t)z * WPERZ  + kPairWoff[PAIR];

  if (ntile < NTILES) {                       // wave-uniform -> EXEC all-ones for WMMA
    int v = vtile*16 + l16;                   // A row M -> (u, v)
    const float* Ra = Rz + (u*64 + v)*NF;     // R[z,u,v,0..NF)
    int n  = ntile*16 + l16;                  // B/D col N -> (i,j) flat
    int ka = hi ? 2 : 0;

    v2f a, b;
    a.x = rload<NF>(Ra, ka);
    a.y = rload<NF>(Ra, ka+1);
    b.x = wload<NF,P>(Wz, ka,   n);
    b.y = wload<NF,P>(Wz, ka+1, n);

    v8f acc = {0.f,0.f,0.f,0.f,0.f,0.f,0.f,0.f};
    acc = __builtin_amdgcn_wmma_f32_16x16x4_f32(false, a, false, b, (short)0, acc, false, false);

    if constexpr (NF > 4) {                   // second K-step (only l2 x l2, NF=5)
      int kb = 4 + ka;
      v2f a2, b2;
      a2.x = rload<NF>(Ra, kb);
      a2.y = rload<NF>(Ra, kb+1);
      b2.x = wload<NF,P>(Wz, kb,   n);
      b2.y = wload<NF,P>(Wz, kb+1, n);
      acc = __builtin_amdgcn_wmma_f32_16x16x4_f32(false, a2, false, b2, (short)0, acc, false, false);
    }

    if (n < P) {                              // D: lanes 0-15 rows r, lanes 16-31 rows r+8
      int vbase = vtile*16 + (hi ? 8 : 0);
      #pragma unroll
      for (int r = 0; r < 8; ++r) lds[vbase + r][n] = acc[r];
    }
  }
  __syncthreads();

  // out[begin_out + u*DIO + i, begin_in + v*DII + j] = lds[v][i*DII + j]
  size_t outBase = (size_t)z*(NOUT*NIN) + (size_t)(kBlkOff[LO] + u*DIO)*NIN + kBlkOff[LI];
  for (int c0 = tid; c0 < CHUNKS; c0 += 256) {
    int t   = c0 << 2;
    int i   = t / ROWLEN;                     // constexpr divisor -> mul/shift
    int rem = t - i*ROWLEN;
    v4f val;
    #pragma unroll
    for (int e = 0; e < 4; ++e) {
      int re = rem + e;
      int vv = re / DII, jj = re - vv*DII;
      val[e] = lds[vv][i*DII + jj];
    }
    // streaming 680MB write-once: bypass L2 residency with NT 128-bit stores
    __builtin_nontemporal_store(val, (v4f*)(out + outBase + (size_t)i*NIN + rem));
  }
}

// ============================ host entry ======================================
extern "C" void kernel_launch(void* const* d_in, const int* in_sizes, int n_in,
                              void* d_out, int out_size, void* d_ws, size_t ws_size,
                              hipStream_t stream) {
  (void)in_sizes; (void)n_in; (void)out_size; (void)ws_size;
  const float* Y  = (const float*)d_in[0];   // [512, 25]
  const float* R  = (const float*)d_in[1];   // [512, 77824]
  const float* nc = (const float*)d_in[2];   // [3, 3]
  float* outF = (float*)d_out;
  float* Cbuf = (float*)d_ws;                // 1225 floats (wigner tensors)
  float* Wbuf = Cbuf + CBUF_FLOATS;          // 512*259 floats

  wigner_init_kernel<<<1, 32, 0, stream>>>(Cbuf);
  w_kernel<<<512, 128, 0, stream>>>(Y, nc, Cbuf, Wbuf);

  dim3 grid(512*64), blk(256);
  tp_kernel<0,0><<<grid, blk, 0, stream>>>(R, Wbuf, outF);
  tp_kernel<0,1><<<grid, blk, 0, stream>>>(R, Wbuf, outF);
  tp_kernel<0,2><<<grid, blk, 0, stream>>>(R, Wbuf, outF);
  tp_kernel<1,0><<<grid, blk, 0, stream>>>(R, Wbuf, outF);
  tp_kernel<1,1><<<grid, blk, 0, stream>>>(R, Wbuf, outF);
  tp_kernel<1,2><<<grid, blk, 0, stream>>>(R, Wbuf, outF);
  tp_kernel<2,0><<<grid, blk, 0, stream>>>(R, Wbuf, outF);
  tp_kernel<2,1><<<grid, blk, 0, stream>>>(R, Wbuf, outF);
  tp_kernel<2,2><<<grid, blk, 0, stream>>>(R, Wbuf, outF);
}